// GCN_6347961663802
// MI455X (gfx1250) — compile-verified
//
#include <hip/hip_runtime.h>
#include <hip/hip_bf16.h>

typedef __attribute__((ext_vector_type(2))) float v2f;
typedef __attribute__((ext_vector_type(8))) float v8f;

// ---------------- degree / norm ----------------
__global__ void k_init_deg(float* deg, int N) {
  int i = blockIdx.x * blockDim.x + threadIdx.x;
  if (i < N) deg[i] = 1.0f;  // self-loop contributes 1 to every node
}

__global__ void k_deg_accum(const long long* ei, float* deg, int E) {
  int e = blockIdx.x * blockDim.x + threadIdx.x;
  if (e < E) atomicAdd(&deg[(int)ei[(long long)E + e]], 1.0f);
}

__global__ void k_dinv(float* deg, int N) {
  int i = blockIdx.x * blockDim.x + threadIdx.x;
  if (i < N) deg[i] = rsqrtf(deg[i]);  // deg >= 1 always (self-loops)
}

// ---------------- utility ----------------
__global__ void k_zero(float* p, long long n) {
  long long i = (long long)blockIdx.x * blockDim.x + threadIdx.x;
  if (i < n) p[i] = 0.0f;
}

// layer1 lift: tmp[i][c] = x[i] * W1[c]   (x is [N,1], W1 is [1,64])
__global__ void k_lin1(const float* x, const float* W1, float* out, long long NF) {
  long long idx = (long long)blockIdx.x * blockDim.x + threadIdx.x;
  if (idx < NF) out[idx] = x[idx >> 6] * W1[idx & 63];
}

// h[i][c] = act(h[i][c] + b[c])
__global__ void k_bias_act(float* buf, const float* b, long long NF, int relu) {
  long long idx = (long long)blockIdx.x * blockDim.x + threadIdx.x;
  if (idx < NF) {
    float v = buf[idx] + b[idx & 63];
    if (relu) v = fmaxf(v, 0.0f);
    buf[idx] = v;
  }
}

// ---------------- edge scatter: out[dst] += tmp[src] * dinv[src]*dinv[dst] ----
// one wave per edge (incl. N synthetic self-loops after the E real edges);
// lane handles features {2l, 2l+1} -> coalesced 256B gather per edge row.
__global__ void __launch_bounds__(256)
k_scatter(const long long* ei, const float* dinv, const float* tmp,
          float* out, int E, int N) {
  int gid  = blockIdx.x * 8 + (threadIdx.x >> 5);
  int lane = threadIdx.x & 31;
  if (gid >= E + N) return;
  int s, d;
  if (gid < E) {
    s = (int)ei[gid];
    d = (int)ei[(long long)E + gid];
  } else {
    s = d = gid - E;
  }
  float nrm = dinv[s] * dinv[d];
  v2f v = *(const v2f*)(tmp + (long long)s * 64 + 2 * lane);
  float* dp = out + (long long)d * 64 + 2 * lane;
  atomicAdd(dp, v.x * nrm);
  atomicAdd(dp + 1, v.y * nrm);
}

// ---------------- in-place GEMM: H[N,64] = H[N,64] @ W[64,64] (f32 WMMA) ----
// Each wave owns a 16-row tile (4 column tiles of 16x16, K=64 as 16 steps of
// V_WMMA_F32_16X16X4_F32). W staged in LDS in K-pair-interleaved form:
//   lW[kp*128 + ((2*col + (k&1) + 32*(kp&1)) & 127)] = W[k][col],  kp = k/2
// so each B fragment {W[kb][col], W[kb+1][col]} is ONE contiguous ds_load_b64
// (no register shuffles), and the 32-float rotation of odd kp rows puts lanes
// 0-15 and 16-31 in disjoint 32-bank halves (conflict-free dual-half reads).
__global__ void __launch_bounds__(256)
k_gemm64_inplace(float* H, const float* W, int N) {
  __shared__ float lW[64 * 64];
  for (int i = threadIdx.x; i < 64 * 64; i += 256) {
    int k = i >> 6, c = i & 63;
    int kp = k >> 1, par = k & 1;
    int fo = (2 * c + par + ((kp & 1) << 5)) & 127;
    lW[kp * 128 + fo] = W[i];
  }
  __syncthreads();

  const int wave = threadIdx.x >> 5;
  const int lane = threadIdx.x & 31;
  const int half = lane >> 4;   // 0: lanes 0-15 (K even pair), 1: lanes 16-31
  const int lr   = lane & 15;
  const int row0 = blockIdx.x * 128 + wave * 16;
  if (row0 + 16 > N) return;    // wave-uniform (EXEC stays all-ones for WMMA)

  // loop-invariant rotated column offsets for the 4 column tiles
  const int rot = half << 5;
  const int fo0 = ((2 * lr) + rot) & 127;
  const int fo1 = (fo0 + 32) & 127;
  const int fo2 = (fo0 + 64) & 127;
  const int fo3 = (fo0 + 96) & 127;

  v8f acc0 = {}, acc1 = {}, acc2 = {}, acc3 = {};
  // A: lane holds row (row0+lr), cols {4k+2*half, 4k+2*half+1}
  const float* Arow = H + (long long)(row0 + lr) * 64 + 2 * half;

#pragma unroll
  for (int kk = 0; kk < 16; ++kk) {
    v2f a = *(const v2f*)(Arow + 4 * kk);
    const float* p = lW + (2 * kk + half) * 128;   // kp = 2*kk + half
    v2f b0 = *(const v2f*)(p + fo0);
    v2f b1 = *(const v2f*)(p + fo1);
    v2f b2 = *(const v2f*)(p + fo2);
    v2f b3 = *(const v2f*)(p + fo3);
    acc0 = __builtin_amdgcn_wmma_f32_16x16x4_f32(false, a, false, b0, (short)0, acc0, false, false);
    acc1 = __builtin_amdgcn_wmma_f32_16x16x4_f32(false, a, false, b1, (short)0, acc1, false, false);
    acc2 = __builtin_amdgcn_wmma_f32_16x16x4_f32(false, a, false, b2, (short)0, acc2, false, false);
    acc3 = __builtin_amdgcn_wmma_f32_16x16x4_f32(false, a, false, b3, (short)0, acc3, false, false);
  }

  // D layout: VGPR v, lane -> row = row0 + v + 8*half, col = 16*j + lr
#pragma unroll
  for (int v = 0; v < 8; ++v) {
    long long r = (long long)(row0 + v + 8 * half) * 64 + lr;
    H[r + 0]  = acc0[v];
    H[r + 16] = acc1[v];
    H[r + 32] = acc2[v];
    H[r + 48] = acc3[v];
  }
}

// ---------------- pooling: pooled[batch[i]] += h[i]; cnt[batch[i]] += 1 ------
__global__ void __launch_bounds__(256)
k_pool(const float* h, const long long* batch, float* pooled, float* cnt, int N) {
  int node = blockIdx.x * 8 + (threadIdx.x >> 5);
  int lane = threadIdx.x & 31;
  if (node >= N) return;
  int g = (int)batch[node];
  v2f v = *(const v2f*)(h + (long long)node * 64 + 2 * lane);
  float* p = pooled + (long long)g * 64 + 2 * lane;
  atomicAdd(p, v.x);
  atomicAdd(p + 1, v.y);
  if (lane == 0) atomicAdd(&cnt[g], 1.0f);
}

// ---------------- head MLP: one thread per graph (G=512, tiny) ---------------
__global__ void k_head(const float* pooled, const float* cnt,
                       const float* dist, const float* sw,
                       const float* Wl, const float* bl,
                       const float* Wl1, const float* bl1,
                       const float* Wl2, const float* bl2,
                       float* out, int G) {
  int g = blockIdx.x * blockDim.x + threadIdx.x;
  if (g >= G) return;
  float inv = 1.0f / fmaxf(cnt[g], 1.0f);
  float a[5];
#pragma unroll
  for (int j = 0; j < 5; ++j) a[j] = bl[j];
  for (int c = 0; c < 64; ++c) {
    float p = pooled[(long long)g * 64 + c] * inv;
#pragma unroll
    for (int j = 0; j < 5; ++j) a[j] += p * Wl[c * 5 + j];
  }
  float in7[7] = { a[0], a[1], a[2], a[3], a[4], dist[g], sw[g] };
  float h2[5];
#pragma unroll
  for (int j = 0; j < 5; ++j) {
    float s = bl1[j];
#pragma unroll
    for (int c = 0; c < 7; ++c) s += in7[c] * Wl1[c * 5 + j];
    h2[j] = fmaxf(s, 0.0f);
  }
  float o = bl2[0];
#pragma unroll
  for (int j = 0; j < 5; ++j) o += h2[j] * Wl2[j];
  out[g] = o;
}

// ---------------- driver ----------------
extern "C" void kernel_launch(void* const* d_in, const int* in_sizes, int n_in,
                              void* d_out, int out_size, void* d_ws, size_t ws_size,
                              hipStream_t stream) {
  const float*     x     = (const float*)d_in[0];
  const long long* ei    = (const long long*)d_in[1];
  const long long* batch = (const long long*)d_in[2];
  const float*     dist  = (const float*)d_in[3];
  const float*     sw    = (const float*)d_in[4];
  const float*     W1    = (const float*)d_in[5];
  const float*     b1    = (const float*)d_in[6];
  const float*     W2    = (const float*)d_in[7];
  const float*     b2    = (const float*)d_in[8];
  const float*     W3    = (const float*)d_in[9];
  const float*     b3    = (const float*)d_in[10];
  const float*     Wl    = (const float*)d_in[11];
  const float*     bl    = (const float*)d_in[12];
  const float*     Wl1   = (const float*)d_in[13];
  const float*     bl1   = (const float*)d_in[14];
  const float*     Wl2   = (const float*)d_in[15];
  const float*     bl2   = (const float*)d_in[16];

  const int N = in_sizes[0];        // x is [N,1]
  const int E = in_sizes[1] / 2;    // edge_index is [2,E]
  const int G = in_sizes[3];        // dist is [G,1]
  const long long NF = (long long)N * 64;

  float* wsf    = (float*)d_ws;
  float* dinv   = wsf;                       // N
  float* bufA   = dinv + N;                  // N*64
  float* bufB   = bufA + (size_t)N * 64;     // N*64
  float* pooled = bufB + (size_t)N * 64;     // G*64
  float* cnt    = pooled + (size_t)G * 64;   // G  (contiguous with pooled)

  const int B = 256;
  auto cdiv = [](long long a, long long b) { return (int)((a + b - 1) / b); };

  // degree + symmetric norm
  k_init_deg<<<cdiv(N, B), B, 0, stream>>>(dinv, N);
  k_deg_accum<<<cdiv(E, B), B, 0, stream>>>(ei, dinv, E);
  k_dinv<<<cdiv(N, B), B, 0, stream>>>(dinv, N);

  // conv1: tmp = x @ W1 ; scatter ; +b1 ; relu
  k_lin1<<<cdiv(NF, B), B, 0, stream>>>(x, W1, bufA, NF);
  k_zero<<<cdiv(NF, B), B, 0, stream>>>(bufB, NF);
  k_scatter<<<cdiv((long long)E + N, 8), B, 0, stream>>>(ei, dinv, bufA, bufB, E, N);
  k_bias_act<<<cdiv(NF, B), B, 0, stream>>>(bufB, b1, NF, 1);

  // conv2: tmp = h @ W2 (WMMA, in-place) ; scatter ; +b2 ; relu
  k_gemm64_inplace<<<cdiv(N, 128), B, 0, stream>>>(bufB, W2, N);
  k_zero<<<cdiv(NF, B), B, 0, stream>>>(bufA, NF);
  k_scatter<<<cdiv((long long)E + N, 8), B, 0, stream>>>(ei, dinv, bufB, bufA, E, N);
  k_bias_act<<<cdiv(NF, B), B, 0, stream>>>(bufA, b2, NF, 1);

  // conv3: tmp = h @ W3 (WMMA, in-place) ; scatter ; +b3 (no relu)
  k_gemm64_inplace<<<cdiv(N, 128), B, 0, stream>>>(bufA, W3, N);
  k_zero<<<cdiv(NF, B), B, 0, stream>>>(bufB, NF);
  k_scatter<<<cdiv((long long)E + N, 8), B, 0, stream>>>(ei, dinv, bufA, bufB, E, N);
  k_bias_act<<<cdiv(NF, B), B, 0, stream>>>(bufB, b3, NF, 0);

  // mean pool + head
  k_zero<<<cdiv((long long)G * 64 + G, B), B, 0, stream>>>(pooled, (long long)G * 64 + G);
  k_pool<<<cdiv(N, 8), B, 0, stream>>>(bufB, batch, pooled, cnt, N);
  k_head<<<cdiv(G, B), B, 0, stream>>>(pooled, cnt, dist, sw,
                                       Wl, bl, Wl1, bl1, Wl2, bl2,
                                       (float*)d_out, G);
}